// GraphDecoder_41016937677244
// MI455X (gfx1250) — compile-verified
//
#include <hip/hip_runtime.h>
#include <math.h>

typedef __attribute__((ext_vector_type(2))) float v2f;
typedef __attribute__((ext_vector_type(8))) float v8f;

#define BINS 16
#define WAVES_PER_BLOCK 8
#define BLOCK_THREADS (WAVES_PER_BLOCK * 32)

// ---------------------------------------------------------------------------
// Zero the 64-float accumulator region in workspace:
//   ws[0..15]  = s_pos,  ws[16..31] = c_pos,
//   ws[32..47] = s_neg,  ws[48..63] = c_neg
// ---------------------------------------------------------------------------
__global__ void gae_init_kernel(float* __restrict__ ws) {
    int i = threadIdx.x;
    if (i < 4 * BINS) ws[i] = 0.0f;
}

// ---------------------------------------------------------------------------
// Per-edge loss with WMMA f32 16x16x4 dot products.
// One wave = 16 edges. A = z[u[tile]] (16xK), B = z[v[tile]]^T (Kx16);
// diagonal of the 16x16 f32 accumulator = per-edge dot products, exact fp32.
// ---------------------------------------------------------------------------
__global__ __launch_bounds__(BLOCK_THREADS) void gae_edge_loss_wmma(
    const float* __restrict__ z,
    const int*   __restrict__ u,
    const int*   __restrict__ v,
    const int*   __restrict__ batch,
    int E, int D,
    float* __restrict__ s_out,   // [BINS] global sum of terms per graph
    float* __restrict__ c_out,   // [BINS] global edge count per graph
    int positive)
{
    __shared__ float lds_s[BINS];
    __shared__ float lds_c[BINS];

    const int tid = threadIdx.x;
    if (tid < BINS) { lds_s[tid] = 0.0f; lds_c[tid] = 0.0f; }
    __syncthreads();

    const int lane = tid & 31;
    const int r    = lane & 15;   // edge slot within the 16-edge tile
    const int half = lane >> 4;   // which K-subpair this lane feeds

    const long wave = (long)blockIdx.x * WAVES_PER_BLOCK + (tid >> 5);
    const long e0   = wave * 16;          // first edge of this wave's tile
    long e = e0 + r;
    const long ec = (e < (long)E) ? e : (long)E - 1;   // clamp, keep EXEC full

    const int ue = u[ec];
    const int ve = v[ec];
    // Both A and B operands load float2 at row + k0 + 2*half (ISA 16x4 / 4x16
    // f32 operand layouts collapse to the same per-lane addressing).
    const float* __restrict__ arow = z + (size_t)ue * (size_t)D + 2 * half;
    const float* __restrict__ brow = z + (size_t)ve * (size_t)D + 2 * half;

    v8f acc = {};
#pragma unroll 8
    for (int k0 = 0; k0 < D; k0 += 4) {
        v2f a = *(const v2f*)(arow + k0);
        v2f b = *(const v2f*)(brow + k0);
        // D = A(16x4) * B(4x16) + C  -> v_wmma_f32_16x16x4_f32
        acc = __builtin_amdgcn_wmma_f32_16x16x4_f32(
            /*neg_a=*/false, a, /*neg_b=*/false, b,
            /*c_mod=*/(short)0, acc, /*reuse_a=*/false, /*reuse_b=*/false);
    }

    // Diagonal extraction: m in [0,8) -> lane m holds acc[m];
    //                      m in [8,16) -> lane m+16 holds acc[m-8].
    const bool isdiag = (lane < 8) || (lane >= 24);
    const int  idx = (lane < 16) ? lane : (lane - 24);
    float dot = acc[0];
    if (idx == 1) dot = acc[1];
    if (idx == 2) dot = acc[2];
    if (idx == 3) dot = acc[3];
    if (idx == 4) dot = acc[4];
    if (idx == 5) dot = acc[5];
    if (idx == 6) dot = acc[6];
    if (idx == 7) dot = acc[7];

    if (isdiag && (e < (long)E)) {
        const int bb = batch[ue] & (BINS - 1);
        const float p = 1.0f / (1.0f + expf(-dot));           // sigmoid
        const float q = positive ? (p + 1e-15f) : ((1.0f - p) + 1e-15f);
        const float term = -logf(q);
        atomicAdd(&lds_s[bb], term);   // ds_add_f32
        atomicAdd(&lds_c[bb], 1.0f);
    }
    __syncthreads();

    if (tid < BINS) {
        atomicAdd(&s_out[tid], lds_s[tid]);   // global_atomic_add_f32
        atomicAdd(&c_out[tid], lds_c[tid]);
    }
}

// ---------------------------------------------------------------------------
// Finalize: out = sum_b( s_pos/max(c_pos,1) + s_neg/max(c_neg,1) ) / B
// ---------------------------------------------------------------------------
__global__ void gae_finalize(const float* __restrict__ ws,
                             const int*   __restrict__ num_batches,
                             float* __restrict__ out)
{
    if (threadIdx.x == 0 && blockIdx.x == 0) {
        int B = num_batches[0];
        if (B > BINS) B = BINS;
        if (B < 1)    B = 1;
        const float* s_pos = ws;
        const float* c_pos = ws + BINS;
        const float* s_neg = ws + 2 * BINS;
        const float* c_neg = ws + 3 * BINS;
        float total = 0.0f;
        for (int b = 0; b < B; ++b) {
            const float cp = (c_pos[b] > 1.0f) ? c_pos[b] : 1.0f;
            const float cn = (c_neg[b] > 1.0f) ? c_neg[b] : 1.0f;
            total += s_pos[b] / cp + s_neg[b] / cn;
        }
        out[0] = total / (float)B;
    }
}

// ---------------------------------------------------------------------------
// Launch: init -> pos edges -> neg edges -> finalize, all on `stream`.
// Inputs (reference order): z[f32 N*D], edge_index[i32 2*E], batch[i32 N],
//                           neg_edge_index[i32 2*E], num_batches[i32 1]
// ---------------------------------------------------------------------------
extern "C" void kernel_launch(void* const* d_in, const int* in_sizes, int n_in,
                              void* d_out, int out_size, void* d_ws, size_t ws_size,
                              hipStream_t stream) {
    (void)n_in; (void)out_size; (void)ws_size;

    const float* z     = (const float*)d_in[0];
    const int*   ei    = (const int*)d_in[1];
    const int*   batch = (const int*)d_in[2];
    const int*   nei   = (const int*)d_in[3];
    const int*   nb    = (const int*)d_in[4];

    const int N = in_sizes[2];
    const int D = in_sizes[0] / N;       // 128
    const int E = in_sizes[1] / 2;       // 800000

    float* ws    = (float*)d_ws;
    float* s_pos = ws;
    float* c_pos = ws + BINS;
    float* s_neg = ws + 2 * BINS;
    float* c_neg = ws + 3 * BINS;

    gae_init_kernel<<<1, 64, 0, stream>>>(ws);

    const long waves  = ((long)E + 15) / 16;
    const int  blocks = (int)((waves + WAVES_PER_BLOCK - 1) / WAVES_PER_BLOCK);

    gae_edge_loss_wmma<<<blocks, BLOCK_THREADS, 0, stream>>>(
        z, ei, ei + E, batch, E, D, s_pos, c_pos, /*positive=*/1);
    gae_edge_loss_wmma<<<blocks, BLOCK_THREADS, 0, stream>>>(
        z, nei, nei + E, batch, E, D, s_neg, c_neg, /*positive=*/0);

    gae_finalize<<<1, 32, 0, stream>>>(ws, nb, (float*)d_out);
}